// CTCCompressor_27058293965444
// MI455X (gfx1250) — compile-verified
//
#include <hip/hip_runtime.h>
#include <hip/hip_bf16.h>
#include <math.h>

// Problem constants (from reference setup_inputs)
#define BB 8
#define TT 2048
#define DD 1024
#define VV 1024
#define BLANK 0
#define EPSW 1e-10f
#define TSHIFT 11   // log2(TT)

typedef __attribute__((ext_vector_type(2))) float v2f;
typedef __attribute__((ext_vector_type(8))) float v8f;
typedef __attribute__((ext_vector_type(4))) unsigned v4u;
typedef __attribute__((ext_vector_type(4))) int v4i;
typedef __attribute__((ext_vector_type(8))) int v8i;

// ---------------------------------------------------------------------------
// Kernel 0: per-row (b,t) softmax stats over V: argmax (first index on ties),
// p = max softmax prob = 1/sum(exp(x - max)); p0[b] = probs[b,0,BLANK].
// One wave32 per row, 8 waves per block, float4 (b128) loads.
// ---------------------------------------------------------------------------
__global__ void __launch_bounds__(256) k_rowstat(const float* __restrict__ logits,
                                                 float* __restrict__ p,
                                                 int* __restrict__ pred,
                                                 float* __restrict__ p0) {
  int row = blockIdx.x * 8 + (threadIdx.x >> 5);
  if (row >= BB * TT) return;
  int lane = threadIdx.x & 31;
  const float* src = logits + (size_t)row * VV;
  const float4* src4 = (const float4*)src;

  float vmax = -3.4e38f;
  int vidx = 0x7fffffff;
#pragma unroll
  for (int j = 0; j < VV / 128; j++) {          // 8 iters, b128 loads
    int e = lane + 32 * j;
    float4 v = src4[e];
    int bi = e * 4;
    if (v.x > vmax || (v.x == vmax && bi + 0 < vidx)) { vmax = v.x; vidx = bi + 0; }
    if (v.y > vmax || (v.y == vmax && bi + 1 < vidx)) { vmax = v.y; vidx = bi + 1; }
    if (v.z > vmax || (v.z == vmax && bi + 2 < vidx)) { vmax = v.z; vidx = bi + 2; }
    if (v.w > vmax || (v.w == vmax && bi + 3 < vidx)) { vmax = v.w; vidx = bi + 3; }
  }
  for (int off = 16; off > 0; off >>= 1) {
    float ov = __shfl_xor(vmax, off, 32);
    int   oi = __shfl_xor(vidx, off, 32);
    if (ov > vmax || (ov == vmax && oi < vidx)) { vmax = ov; vidx = oi; }
  }
  float s = 0.0f;
#pragma unroll
  for (int j = 0; j < VV / 128; j++) {
    float4 v = src4[lane + 32 * j];
    s += __expf(v.x - vmax) + __expf(v.y - vmax) +
         __expf(v.z - vmax) + __expf(v.w - vmax);
  }
  for (int off = 16; off > 0; off >>= 1) s += __shfl_xor(s, off, 32);

  if (lane == 0) {
    p[row] = 1.0f / s;          // probs at argmax
    pred[row] = vidx;
    if ((row & (TT - 1)) == 0) {
      int b = row >> TSHIFT;
      p0[b] = __expf(src[BLANK] - vmax) / s;
    }
  }
}

// ---------------------------------------------------------------------------
// Kernel 1: per-batch segment scan. One block (256 thr) per batch, 8 frames
// per thread. Produces seg_of_t (-1 if frame not in a segment), seg_start[s],
// n_info[b] = {n_seg, n_eff, fallback}, and new_lengths tail of d_out.
// ---------------------------------------------------------------------------
__global__ void __launch_bounds__(256) k_scan(const int* __restrict__ pred,
                                              const int* __restrict__ lengths,
                                              int* __restrict__ seg_of_t,
                                              int* __restrict__ seg_start,
                                              int* __restrict__ n_info,
                                              float* __restrict__ out_lengths) {
  int b = blockIdx.x;
  int tid = threadIdx.x;
  __shared__ int s_cnt[256];

  int len = lengths[b];
  const int* predb = pred + (size_t)b * TT;

  int base = tid * 8;
  int nb[8];
  int cnt = 0;
#pragma unroll
  for (int i = 0; i < 8; i++) {
    int t = base + i;
    int pr = predb[t];
    int pv = (t == 0) ? -1 : predb[t - 1];
    bool valid = t < len;
    bool bnd = valid && (pr != pv) && (pr != BLANK);
    nb[i] = bnd ? 1 : 0;
    cnt += nb[i];
  }
  s_cnt[tid] = cnt;
  __syncthreads();
  for (int off = 1; off < 256; off <<= 1) {
    int v = s_cnt[tid];
    int add = (tid >= off) ? s_cnt[tid - off] : 0;
    __syncthreads();
    s_cnt[tid] = v + add;
    __syncthreads();
  }
  int excl = s_cnt[tid] - cnt;
  int total = s_cnt[255];

  int run = excl;
#pragma unroll
  for (int i = 0; i < 8; i++) {
    int t = base + i;
    run += nb[i];
    int sidx = run - 1;
    int pr = predb[t];
    bool fis = (t < len) && (pr != BLANK);
    seg_of_t[(size_t)b * TT + t] = (fis && sidx >= 0) ? sidx : -1;
    if (nb[i]) seg_start[(size_t)b * TT + sidx] = t;
  }
  if (tid == 0) {
    int neff = total > 0 ? total : 1;
    n_info[b * 3 + 0] = total;
    n_info[b * 3 + 1] = neff;
    n_info[b * 3 + 2] = (total == 0) ? 1 : 0;
    if (total == 0) seg_start[(size_t)b * TT + 0] = 0;
    out_lengths[b] = (float)neff;     // new_lengths (tail of d_out)
  }
}

// ---------------------------------------------------------------------------
// Kernel 2: deterministic per-segment sums of p (no float atomics so graph
// replays are bit-stable). One block per batch.
// ---------------------------------------------------------------------------
__global__ void __launch_bounds__(256) k_segsum(const float* __restrict__ p,
                                                const int* __restrict__ seg_of_t,
                                                const int* __restrict__ seg_start,
                                                const int* __restrict__ n_info,
                                                const int* __restrict__ lengths,
                                                float* __restrict__ seg_sum) {
  int b = blockIdx.x;
  int n_seg = n_info[b * 3 + 0];
  int len = lengths[b];
  for (int s = threadIdx.x; s < n_seg; s += 256) {
    int tb = seg_start[(size_t)b * TT + s];
    int te = (s + 1 < n_seg) ? seg_start[(size_t)b * TT + s + 1] : len;
    float acc = 0.0f;
    for (int t = tb; t < te; t++) {
      if (seg_of_t[(size_t)b * TT + t] == s) acc += p[(size_t)b * TT + t];
    }
    seg_sum[(size_t)b * TT + s] = acc;
  }
}

// ---------------------------------------------------------------------------
// Kernel 3: packed per-frame (seg, weight) pairs: swg[b*T+t] = {seg, w_bits}
// with w = p / (seg_sum + eps); fallback at (t=0,s=0).
// ---------------------------------------------------------------------------
__global__ void __launch_bounds__(256) k_weights(const float* __restrict__ p,
                                                 const float* __restrict__ p0,
                                                 const float* __restrict__ seg_sum,
                                                 const int* __restrict__ seg_of_t,
                                                 const int* __restrict__ n_info,
                                                 uint2* __restrict__ swg) {
  int i = blockIdx.x * 256 + threadIdx.x;
  if (i >= BB * TT) return;
  int b = i >> TSHIFT;
  int t = i & (TT - 1);
  int s = seg_of_t[i];
  float wv = 0.0f;
  if (s >= 0) wv = p[i] / (seg_sum[(size_t)b * TT + s] + EPSW);
  if (t == 0 && n_info[b * 3 + 2]) {   // n_seg==0 fallback
    s = 0;
    float q = p0[b];
    wv = q / (q + EPSW);
  }
  uint2 e;
  e.x = (unsigned)s;
  e.y = __float_as_uint(wv);
  swg[i] = e;
}

// ---------------------------------------------------------------------------
// Zero-fill the (B,T,D) compressed output (rows >= n_seg must be exact zero).
// ---------------------------------------------------------------------------
__global__ void __launch_bounds__(256) k_zero(float4* __restrict__ out, size_t n4) {
  size_t i = (size_t)blockIdx.x * 256 + threadIdx.x;
  size_t stride = (size_t)gridDim.x * 256;
  float4 z = make_float4(0.f, 0.f, 0.f, 0.f);
  for (; i < n4; i += stride) out[i] = z;
}

// ---------------------------------------------------------------------------
// Kernel 4: WMMA segmented reduction.
//   compressed[b, s0:s0+16, :] = A(16 seg x K frames) * B(K frames x D)
// via V_WMMA_F32_16X16X4_F32. The (seg,w) frame window is DMA'd into LDS by
// the Tensor Data Mover (TENSOR_LOAD_TO_LDS, 1-D tile of win x 8B), synced
// with s_wait_tensorcnt. Main loop clamp-free + 2x unrolled with prefetch;
// one peeled tail chunk uses clamps + LDS padding.
// ---------------------------------------------------------------------------
__global__ void __launch_bounds__(256) k_wmma(const float* __restrict__ hidden,
                                              const uint2* __restrict__ swg,
                                              const int* __restrict__ seg_start,
                                              const int* __restrict__ n_info,
                                              const int* __restrict__ lengths,
                                              float* __restrict__ out) {
  int b = blockIdx.y;
  int s0 = blockIdx.x * 16;
  int n_seg = n_info[b * 3 + 0];
  int n_eff = n_info[b * 3 + 1];
  if (s0 >= n_eff) return;   // rows already zeroed by k_zero
  int len = lengths[b];

  int t_begin = seg_start[(size_t)b * TT + s0];
  int t_end = (s0 + 16 < n_seg) ? seg_start[(size_t)b * TT + s0 + 16] : len;
  int win = t_end - t_begin;
  if (win < 0) win = 0;

  // (seg, w_bits) interleaved pairs; one b128 LDS load covers 2 frames.
  __shared__ __align__(16) uint2 s_sw[TT + 8];

  int lane = threadIdx.x & 31;
  int wave = threadIdx.x >> 5;

  // --- TDM: DMA the packed (seg,w) window global -> LDS (wave 0 only) ---
  if (wave == 0 && win > 0) {
    unsigned lds_addr = (unsigned)(unsigned long long)&s_sw[0];  // flat low 32 = LDS offset
    unsigned long long gaddr =
        (unsigned long long)(const void*)(swg + ((size_t)b * TT + t_begin));
    unsigned uwin = (unsigned)win;
    // D# group 0: count=1 | lds_addr | global_addr[56:0] | type=2
    v4u g0;
    g0.x = 1u;                                   // count=1, user mode
    g0.y = lds_addr;                             // LDS byte address
    g0.z = (unsigned)(gaddr & 0xFFFFFFFFu);      // global_addr[31:0]
    g0.w = (unsigned)((gaddr >> 32) & 0x01FFFFFFu) | (2u << 30);  // [56:32] | type=2
    // D# group 1: data_size=8B, 1-D tile: tensor_dim0=tile_dim0=win
    v8i g1;
    g1[0] = (int)(3u << 16);                     // workgroup_mask=0, data_size=3 (8B)
    g1[1] = (int)((uwin & 0xFFFFu) << 16);       // tensor_dim0[15:0] at bits 63:48
    g1[2] = (int)((uwin >> 16) | (1u << 16));    // tensor_dim0[31:16] | tensor_dim1=1
    g1[3] = (int)((uwin & 0xFFFFu) << 16);       // tensor_dim1[31:16]=0 | tile_dim0=win
    g1[4] = 0;                                   // tile_dim1=0, tile_dim2=0 (unused)
    g1[5] = (int)uwin;                           // tensor_dim0_stride[31:0]
    g1[6] = 0;                                   // stride[47:32]=0, dim1_stride[15:0]=0
    g1[7] = 0;
    v4i gz4 = {0, 0, 0, 0};                      // groups 2/3 unused (<=2D)
    v8i gz8 = {0, 0, 0, 0, 0, 0, 0, 0};          // extra descriptor group (unused)
    __builtin_amdgcn_tensor_load_to_lds(g0, g1, gz4, gz4, gz8, 0);
  }
  // pad entries [win, win+8): seg=-1, w=0 (normal DS stores)
  if (threadIdx.x < 8) {
    uint2 e; e.x = 0xFFFFFFFFu; e.y = 0u;
    s_sw[win + threadIdx.x] = e;
  }
  if (wave == 0) __builtin_amdgcn_s_wait_tensorcnt(0);
  __syncthreads();

  int m  = lane & 15;                 // A row / B col / C col
  int k0 = (lane >> 4) << 1;          // 0 for lanes 0-15, 2 for lanes 16-31
  int sA = s0 + m;
  const float* hb = hidden + (size_t)b * TT * DD;
  int winMain = win & ~3;             // full chunks: provably in-bounds

  for (int j = 0; j < 8; j++) {
    int d0 = (wave * 8 + j) * 16;     // 64 d-tiles split across 8 waves
    v8f c = {0.f, 0.f, 0.f, 0.f, 0.f, 0.f, 0.f, 0.f};

    // lane's column pointer at frame (t_begin + k0)
    const float* pB = hb + ((size_t)(t_begin + k0)) * DD + d0 + m;

    int t = 0;
    // ---- main loop, 2 chunks (K=8 frames) per iteration, clamp-free ----
    for (; t + 8 <= winMain; t += 8) {
      const float* q0 = pB + (size_t)t * DD;
      const float* q1 = q0 + 4 * DD;
      // prefetch next unrolled iteration's rows
      __builtin_prefetch(q0 + 8 * DD, 0, 1);
      __builtin_prefetch(q0 + 9 * DD, 0, 1);
      float b00 = q0[0], b01 = q0[DD];
      float b10 = q1[0], b11 = q1[DD];
      uint4 e0 = *(const uint4*)&s_sw[t + k0];
      uint4 e1 = *(const uint4*)&s_sw[t + 4 + k0];
      v2f a0, a1, bm0, bm1;
      a0.x = ((int)e0.x == sA) ? __uint_as_float(e0.y) : 0.0f;
      a0.y = ((int)e0.z == sA) ? __uint_as_float(e0.w) : 0.0f;
      a1.x = ((int)e1.x == sA) ? __uint_as_float(e1.y) : 0.0f;
      a1.y = ((int)e1.z == sA) ? __uint_as_float(e1.w) : 0.0f;
      bm0.x = b00; bm0.y = b01;
      bm1.x = b10; bm1.y = b11;
      c = __builtin_amdgcn_wmma_f32_16x16x4_f32(false, a0, false, bm0, (short)0, c, false, false);
      c = __builtin_amdgcn_wmma_f32_16x16x4_f32(false, a1, false, bm1, (short)0, c, false, false);
    }
    // ---- leftover single full chunk ----
    for (; t < winMain; t += 4) {
      const float* q0 = pB + (size_t)t * DD;
      float b00 = q0[0], b01 = q0[DD];
      uint4 e0 = *(const uint4*)&s_sw[t + k0];
      v2f a0, bm0;
      a0.x = ((int)e0.x == sA) ? __uint_as_float(e0.y) : 0.0f;
      a0.y = ((int)e0.z == sA) ? __uint_as_float(e0.w) : 0.0f;
      bm0.x = b00; bm0.y = b01;
      c = __builtin_amdgcn_wmma_f32_16x16x4_f32(false, a0, false, bm0, (short)0, c, false, false);
    }
    // ---- tail partial chunk: A masked by LDS padding, clamped B address ----
    if (win & 3) {
      int gt0 = t_begin + winMain + k0;     if (gt0 > TT - 1) gt0 = TT - 1;
      int gt1 = t_begin + winMain + k0 + 1; if (gt1 > TT - 1) gt1 = TT - 1;
      uint4 e0 = *(const uint4*)&s_sw[winMain + k0];
      v2f a0, bm0;
      a0.x = ((int)e0.x == sA) ? __uint_as_float(e0.y) : 0.0f;
      a0.y = ((int)e0.z == sA) ? __uint_as_float(e0.w) : 0.0f;
      bm0.x = hb[(size_t)gt0 * DD + d0 + m];
      bm0.y = hb[(size_t)gt1 * DD + d0 + m];
      c = __builtin_amdgcn_wmma_f32_16x16x4_f32(false, a0, false, bm0, (short)0, c, false, false);
    }

    // C/D layout: VGPR r -> M = r (lanes 0-15) or r+8 (lanes 16-31), N = lane&15
    int sBase = s0 + ((lane >> 4) << 3);
    float* ob = out + ((size_t)b * TT + sBase) * DD + d0 + m;
#pragma unroll
    for (int r = 0; r < 8; r++) ob[(size_t)r * DD] = c[r];
  }
}

// ---------------------------------------------------------------------------
extern "C" void kernel_launch(void* const* d_in, const int* in_sizes, int n_in,
                              void* d_out, int out_size, void* d_ws, size_t ws_size,
                              hipStream_t stream) {
  const float* hidden  = (const float*)d_in[0];
  const float* logits  = (const float*)d_in[1];
  const int*   lengths = (const int*)d_in[2];
  float* out = (float*)d_out;
  float* out_lengths = out + (size_t)BB * TT * DD;  // new_lengths tail

  // Workspace carve-up (~460 KB total)
  char* ws = (char*)d_ws;
  size_t off = 0;
  auto carve = [&](size_t bytes) -> void* {
    void* ptr = ws + off;
    off = (off + bytes + 255) & ~(size_t)255;
    return ptr;
  };
  int*   pred     = (int*)  carve((size_t)BB * TT * sizeof(int));
  float* p        = (float*)carve((size_t)BB * TT * sizeof(float));
  float* p0       = (float*)carve((size_t)BB * sizeof(float));
  int*   seg_of_t = (int*)  carve((size_t)BB * TT * sizeof(int));
  int*   seg_strt = (int*)  carve((size_t)BB * TT * sizeof(int));
  float* seg_sum  = (float*)carve((size_t)BB * TT * sizeof(float));
  int*   n_info   = (int*)  carve((size_t)BB * 3 * sizeof(int));
  uint2* swg      = (uint2*)carve((size_t)BB * TT * sizeof(uint2));
  (void)in_sizes; (void)n_in; (void)out_size; (void)ws_size;

  // 1) softmax stats: 16384 rows, 8 waves per 256-thread block
  k_rowstat<<<(BB * TT) / 8, 256, 0, stream>>>(logits, p, pred, p0);
  // 2) segment scan: one block per batch
  k_scan<<<BB, 256, 0, stream>>>(pred, lengths, seg_of_t, seg_strt, n_info, out_lengths);
  // 3) deterministic per-segment sums
  k_segsum<<<BB, 256, 0, stream>>>(p, seg_of_t, seg_strt, n_info, lengths, seg_sum);
  // 4) packed (seg, weight) pairs + fallback
  k_weights<<<(BB * TT) / 256, 256, 0, stream>>>(p, p0, seg_sum, seg_of_t, n_info, swg);
  // 5) zero output, then TDM + WMMA segmented reduction
  k_zero<<<4096, 256, 0, stream>>>((float4*)out, (size_t)BB * TT * DD / 4);
  dim3 grid(TT / 16, BB);
  k_wmma<<<grid, 256, 0, stream>>>(hidden, swg, seg_strt, n_info, lengths, out);
}